// LinearStitcher_12025908428992
// MI455X (gfx1250) — compile-verified
//
#include <hip/hip_runtime.h>

// ---------------------------------------------------------------------------
// LinearStitcher on gfx1250 (MI455X, wave32, WMMA)
//
// out[b,t, a*128 + c] = sum_k x[b,t, idx[a][k]] * W[a,k,c] + bias[a,c]
//   A=10 areas, NPA=192 neurons/area, C=128, M = B*T = 32768 rows.
//
// Memory-bound (~38 FLOP/byte => ~18us floor at 23.3 TB/s): keep fp32
// end-to-end and use V_WMMA_F32_16X16X4_F32; optimize HBM/L2 traffic and
// the inner-loop instruction mix. W is pre-swizzled in LDS into the exact
// per-lane WMMA B-operand layout so each B fragment is one ds_load_b64
// into an even-aligned VGPR pair (no 2addr loads, no mov shuffling).
// ---------------------------------------------------------------------------

typedef float v2f __attribute__((ext_vector_type(2)));
typedef float v8f __attribute__((ext_vector_type(8)));

#define NA     10
#define NPA    192
#define NC     128
#define NN     (NA * NPA)    // 1920 neurons total
#define OSTR   (NA * NC)     // 1280 output row stride
#define MT     128           // output rows per block
#define KSTEPS (NPA / 4)     // 48 K-steps of 4

// --------------------------------------------------------------------------
// Build gather table: idx[a*NPA + r] = r-th (ascending) position i with
// labels[i] == a.  Rank-by-scan over LDS-cached labels; one block.
// --------------------------------------------------------------------------
__global__ __launch_bounds__(256)
void build_idx_kernel(const int* __restrict__ labels, int* __restrict__ idx) {
    __shared__ int lbl[NN];
    for (int i = threadIdx.x; i < NN; i += 256) lbl[i] = labels[i];
    __syncthreads();
    for (int i = threadIdx.x; i < NN; i += 256) {
        const int a = lbl[i];
        int r = 0;
        for (int j = 0; j < i; ++j) r += (lbl[j] == a) ? 1 : 0;
        idx[a * NPA + r] = i;
    }
}

// --------------------------------------------------------------------------
// Main GEMM: grid = (area, rowTile). Block = 256 threads = 8 waves.
// Each block: 128 rows x 128 cols of one area. Each wave: 16 rows, 8
// column sub-tiles of 16 -> 8 x v8f accumulators, K-loop in steps of 4
// feeding v_wmma_f32_16x16x4_f32.
//
// LDS B-operand swizzle: for K-step kk, subtile n, lane L the B fragment
// VGPR pair is { W[kk*4 + 2*(L>=16) + 0][n*16 + (L&15)],
//                W[kk*4 + 2*(L>=16) + 1][n*16 + (L&15)] }   (K / K+1 rows,
// lanes 16-31 take K+2 / K+3), stored contiguously per lane:
//   sWB[(((kk*8)+n)*32 + L)*2 + {0,1}]
// => one aligned ds_load_b64 per fragment, 256B contiguous per (kk,n).
// --------------------------------------------------------------------------
__global__ __launch_bounds__(256)
void stitch_wmma_kernel(const float* __restrict__ x,
                        const float* __restrict__ W,
                        const float* __restrict__ bias,
                        const int*   __restrict__ idx,
                        float*       __restrict__ out) {
    __shared__ __attribute__((aligned(16))) float sWB[KSTEPS * 8 * 64]; // 96 KB
    __shared__ __attribute__((aligned(8)))  int   sIdx[NPA];

    const int tid     = threadIdx.x;
    const int area    = blockIdx.x;       // areas adjacent in launch order ->
    const int rowBase = blockIdx.y * MT;  // same x rows stay hot in L2

    // ---- Stage W[area] (192x128 f32) into swizzled LDS + gather indices ----
    {
        const float* wa = W + area * NPA * NC;
        for (int i = tid; i < KSTEPS * 8 * 32; i += 256) {
            const int lane_ = i & 31;
            const int n_    = (i >> 5) & 7;
            const int kk_   = i >> 8;
            const int krow  = kk_ * 4 + ((lane_ >> 4) << 1);
            const int col   = n_ * 16 + (lane_ & 15);
            v2f v;
            v.x = wa[krow * NC + col];
            v.y = wa[(krow + 1) * NC + col];
            *(v2f*)(&sWB[i * 2]) = v;
        }
        for (int i = tid; i < NPA; i += 256) sIdx[i] = idx[area * NPA + i];
    }
    __syncthreads();

    const int lane = tid & 31;
    const int wave = tid >> 5;
    const int ncol = lane & 15;            // N position within 16-wide subtile
    const int koff = (lane >> 4) << 1;     // lanes 0-15: K,K+1 ; 16-31: K+2,K+3
    const int mrow = rowBase + wave * 16 + ncol;   // A-operand row for this lane
    const float* __restrict__ xrow  = x + (long)mrow * NN;
    const float* __restrict__ wbase = sWB + lane * 2;

    v8f acc[8];
#pragma unroll
    for (int n = 0; n < 8; ++n) acc[n] = (v8f){0.f,0.f,0.f,0.f,0.f,0.f,0.f,0.f};

    for (int kk = 0; kk < KSTEPS; ++kk) {
        // Two consecutive gather indices as one aligned ds_load_b64.
        const int2 ii = *(const int2*)(&sIdx[kk * 4 + koff]);
        // A fragment (16x4 f32): per-lane gathered x loads.
        v2f af;
        af.x = xrow[ii.x];
        af.y = xrow[ii.y];

#pragma unroll
        for (int n = 0; n < 8; ++n) {
            // B fragment: single aligned ds_load_b64 (n*256B immediate offset).
            const v2f bf = *(const v2f*)(wbase + (kk * 8 + n) * 64);
            acc[n] = __builtin_amdgcn_wmma_f32_16x16x4_f32(
                false, af, false, bf, (short)0, acc[n], false, false);
        }
    }

    // Bias add + store. C/D layout: VGPR r -> M = r + 8*(lane>=16), N = lane&15.
    const int rhalf = (lane >> 4) * 8;
#pragma unroll
    for (int n = 0; n < 8; ++n) {
        const int col = n * 16 + ncol;
        const float bv = bias[area * NC + col];
#pragma unroll
        for (int r = 0; r < 8; ++r) {
            const int row = rowBase + wave * 16 + rhalf + r;
            out[(long)row * OSTR + area * NC + col] = acc[n][r] + bv;
        }
    }
}

// --------------------------------------------------------------------------
// Host-side launcher.
// Inputs (setup_inputs order): 0=x f32 (B,T,N), 1=W f32 (A,NPA,C),
// 2=b f32 (A,C), 3=neuron_regions i32 (B,N), 4=is_left (unused).
// --------------------------------------------------------------------------
extern "C" void kernel_launch(void* const* d_in, const int* in_sizes, int n_in,
                              void* d_out, int out_size, void* d_ws, size_t ws_size,
                              hipStream_t stream) {
    const float* x    = (const float*)d_in[0];
    const float* W    = (const float*)d_in[1];
    const float* bias = (const float*)d_in[2];
    const int*   nr   = (const int*)d_in[3];   // use row 0 (all rows identical)

    int* idx = (int*)d_ws;                     // 10*192 ints = 7680 B scratch

    const int Mrows  = in_sizes[0] / NN;       // B*T = 32768
    const int mTiles = Mrows / MT;             // 256

    build_idx_kernel<<<1, 256, 0, stream>>>(nr, idx);
    stitch_wmma_kernel<<<dim3(NA, mTiles), 256, 0, stream>>>(x, W, bias, idx,
                                                             (float*)d_out);
}